// VGAELinkPredictor_77481210020191
// MI455X (gfx1250) — compile-verified
//
#include <hip/hip_runtime.h>
#include <hip/hip_bf16.h>
#include <cmath>

typedef float v2f __attribute__((ext_vector_type(2)));
typedef float v8f __attribute__((ext_vector_type(8)));

// ---------------- degree / normalization ----------------

__global__ __launch_bounds__(256) void k_deg_init(unsigned int* __restrict__ deg, int n) {
    int i = blockIdx.x * blockDim.x + threadIdx.x;
    if (i < n) deg[i] = 1u;   // self-loop contributes 1 to in-degree
}

__global__ __launch_bounds__(256) void k_deg_edges(const int* __restrict__ dst,
                                                   unsigned int* __restrict__ deg, int e) {
    int i = blockIdx.x * blockDim.x + threadIdx.x;
    if (i < e) atomicAdd(&deg[dst[i]], 1u);
}

__global__ __launch_bounds__(256) void k_dinv(const unsigned int* __restrict__ deg,
                                              float* __restrict__ dinv, int n) {
    int i = blockIdx.x * blockDim.x + threadIdx.x;
    if (i < n) {
        unsigned int d = deg[i];
        dinv[i] = rsqrtf((float)(d < 1u ? 1u : d));
    }
}

// ---------------- fp32 WMMA GEMM:  Y[M,Ncols] = X[M,K] @ W[K,Ncols] ----------------
// One wave computes a 16x64 output strip: 4 accumulators share one A fragment
// per k-step (4 V_WMMA_F32_16X16X4_F32 per A load -> 4x less A re-fetch).
// blockDim = (32, Ncols/64).
// A 16x4 frag: lane l holds M = l&15, K = r + 2*(l>>4) in vgpr r (r=0,1)
// B 4x16 frag: lane l holds N = l&15, K = r + 2*(l>>4) in vgpr r
// C/D 16x16:  vgpr r, lane l -> row = r + 8*(l>>4), col = l&15

__global__ __launch_bounds__(64) void k_gemm_wmma_f32(const float* __restrict__ X,
                                                      const float* __restrict__ W,
                                                      float* __restrict__ Y,
                                                      int K, int Ncols) {
    const int lane = threadIdx.x;          // 0..31
    const int half = lane >> 4;            // 0 or 1
    const int lm   = lane & 15;
    const int n0   = threadIdx.y * 64;     // 64-wide column strip
    const int m0   = blockIdx.x * 16;      // row tile (M divisible by 16)

    const float* __restrict__ xrow  = X + (size_t)(m0 + lm) * (size_t)K + 2 * half;
    const float* __restrict__ wbase = W + (size_t)(2 * half) * (size_t)Ncols + n0 + lm;

    v8f c0 = {}, c1 = {}, c2 = {}, c3 = {};
    for (int k0 = 0; k0 < K; k0 += 4) {
        v2f a = *(const v2f*)(xrow + k0);                   // contiguous -> b64 load
        const float* wk = wbase + (size_t)k0 * (size_t)Ncols;
        v2f b0, b1, b2, b3;
        b0.x = wk[0];  b0.y = wk[0 + Ncols];
        b1.x = wk[16]; b1.y = wk[16 + Ncols];
        b2.x = wk[32]; b2.y = wk[32 + Ncols];
        b3.x = wk[48]; b3.y = wk[48 + Ncols];
        c0 = __builtin_amdgcn_wmma_f32_16x16x4_f32(false, a, false, b0, (short)0, c0, false, false);
        c1 = __builtin_amdgcn_wmma_f32_16x16x4_f32(false, a, false, b1, (short)0, c1, false, false);
        c2 = __builtin_amdgcn_wmma_f32_16x16x4_f32(false, a, false, b2, (short)0, c2, false, false);
        c3 = __builtin_amdgcn_wmma_f32_16x16x4_f32(false, a, false, b3, (short)0, c3, false, false);
    }

#pragma unroll
    for (int r = 0; r < 8; ++r) {
        int row = m0 + r + 8 * half;
        float* yb = Y + (size_t)row * (size_t)Ncols + n0 + lm;
        yb[0]  = c0[r];
        yb[16] = c1[r];
        yb[32] = c2[r];
        yb[48] = c3[r];
    }
}

// ---------------- self-loop + bias initialization: out = b + xw * dinv^2 ----------------

__global__ __launch_bounds__(256) void k_selfloop_bias(const float* __restrict__ xw,
                                                       const float* __restrict__ dinv,
                                                       const float* __restrict__ bias,
                                                       float* __restrict__ out,
                                                       int n, int d) {
    int i = blockIdx.x * blockDim.x + threadIdx.x;
    int total = n * d;
    if (i < total) {
        int node = i / d;
        int col  = i % d;
        float di = dinv[node];
        out[i] = bias[col] + xw[i] * di * di;
    }
}

// ---------------- edge scatter-add (wave32 per edge), 128-wide: float4 per lane ----------------

__global__ __launch_bounds__(256) void k_scatter128(const int* __restrict__ src,
                                                    const int* __restrict__ dst,
                                                    const float* __restrict__ xw,
                                                    const float* __restrict__ dinv,
                                                    float* __restrict__ out, int e) {
    int lane = threadIdx.x & 31;
    int ew = (int)((blockIdx.x * blockDim.x + threadIdx.x) >> 5);  // one wave per edge
    if (ew >= e) return;
    int s = src[ew];
    int d = dst[ew];
    float coef = dinv[s] * dinv[d];
    const float4* xs = (const float4*)(xw + (size_t)s * 128);
    float4 v = xs[lane];
    float* od = out + (size_t)d * 128 + lane * 4;
    unsafeAtomicAdd(od + 0, v.x * coef);
    unsafeAtomicAdd(od + 1, v.y * coef);
    unsafeAtomicAdd(od + 2, v.z * coef);
    unsafeAtomicAdd(od + 3, v.w * coef);
}

// ---------------- edge scatter-add, 64-wide: float2 per lane ----------------

__global__ __launch_bounds__(256) void k_scatter64(const int* __restrict__ src,
                                                   const int* __restrict__ dst,
                                                   const float* __restrict__ xw,
                                                   const float* __restrict__ dinv,
                                                   float* __restrict__ out, int e) {
    int lane = threadIdx.x & 31;
    int ew = (int)((blockIdx.x * blockDim.x + threadIdx.x) >> 5);
    if (ew >= e) return;
    int s = src[ew];
    int d = dst[ew];
    float coef = dinv[s] * dinv[d];
    const float2* xs = (const float2*)(xw + (size_t)s * 64);
    float2 v = xs[lane];
    float* od = out + (size_t)d * 64 + lane * 2;
    unsafeAtomicAdd(od + 0, v.x * coef);
    unsafeAtomicAdd(od + 1, v.y * coef);
}

// ---------------- ReLU ----------------

__global__ __launch_bounds__(256) void k_relu(float* __restrict__ a, int total) {
    int i = blockIdx.x * blockDim.x + threadIdx.x;
    if (i < total) a[i] = fmaxf(a[i], 0.0f);
}

// ---------------- head: sigmoid(z @ fcW + fcb), wave per node over 64 features ----------------

__global__ __launch_bounds__(256) void k_head(const float* __restrict__ z,
                                              const float* __restrict__ fcW,
                                              const float* __restrict__ fcb,
                                              float* __restrict__ out, int n) {
    int lane = threadIdx.x & 31;
    int node = (int)((blockIdx.x * blockDim.x + threadIdx.x) >> 5);
    if (node >= n) return;
    const float* zr = z + (size_t)node * 64;
    float s = zr[lane] * fcW[lane] + zr[lane + 32] * fcW[lane + 32];
#pragma unroll
    for (int off = 16; off > 0; off >>= 1) s += __shfl_down(s, off, 32);
    if (lane == 0) out[node] = 1.0f / (1.0f + expf(-(s + fcb[0])));
}

// ---------------- launcher ----------------

extern "C" void kernel_launch(void* const* d_in, const int* in_sizes, int n_in,
                              void* d_out, int out_size, void* d_ws, size_t ws_size,
                              hipStream_t stream) {
    const int DIN  = 128;
    const int DHID = 128;
    const int DOUT = 64;

    const float* x   = (const float*)d_in[0];
    const int*   ei  = (const int*)d_in[1];
    const float* W1  = (const float*)d_in[2];
    const float* b1  = (const float*)d_in[3];
    const float* W2  = (const float*)d_in[4];
    const float* b2  = (const float*)d_in[5];
    const float* fcW = (const float*)d_in[6];
    const float* fcb = (const float*)d_in[7];
    float* out = (float*)d_out;

    const int N = in_sizes[0] / DIN;       // 50000
    const int E = in_sizes[1] / 2;         // 800000
    const int* src = ei;
    const int* dst = ei + E;

    // workspace layout (floats):
    //   [0, N)            deg (as uint)
    //   [N, 2N)           dinv
    //   [2N, 2N+128N)     bufA: xw1; later reused as hw (first 64N) and z (second 64N)
    //   [2N+128N, ...)    h (128N)
    float* ws = (float*)d_ws;
    unsigned int* deg = (unsigned int*)ws;
    float* dinv = ws + (size_t)N;
    float* bufA = ws + (size_t)2 * N;               // N*128
    float* h    = bufA + (size_t)N * DHID;          // N*128
    float* hw   = bufA;                             // N*64  (xw1 dead by then)
    float* z    = bufA + (size_t)N * DOUT;          // N*64

    const int T = 256;

    // degree + normalization
    k_deg_init<<<(N + T - 1) / T, T, 0, stream>>>(deg, N);
    k_deg_edges<<<(E + T - 1) / T, T, 0, stream>>>(dst, deg, E);
    k_dinv<<<(N + T - 1) / T, T, 0, stream>>>(deg, dinv, N);

    // layer 1: xw1 = x @ W1   (WMMA fp32, 16x64 strip per wave)
    {
        dim3 blk(32, DHID / 64);                    // 2 waves cover 128 cols
        dim3 grd(N / 16);                           // 50000 / 16 = 3125 exact
        k_gemm_wmma_f32<<<grd, blk, 0, stream>>>(x, W1, bufA, DIN, DHID);
    }
    k_selfloop_bias<<<((N * DHID) + T - 1) / T, T, 0, stream>>>(bufA, dinv, b1, h, N, DHID);
    {
        long long threads = (long long)E * 32;
        k_scatter128<<<(int)((threads + T - 1) / T), T, 0, stream>>>(src, dst, bufA, dinv, h, E);
    }
    k_relu<<<((N * DHID) + T - 1) / T, T, 0, stream>>>(h, N * DHID);

    // layer 2: hw = h @ W2   (WMMA fp32)
    {
        dim3 blk(32, DOUT / 64);                    // 1 wave covers 64 cols
        dim3 grd(N / 16);
        k_gemm_wmma_f32<<<grd, blk, 0, stream>>>(h, W2, hw, DHID, DOUT);
    }
    k_selfloop_bias<<<((N * DOUT) + T - 1) / T, T, 0, stream>>>(hw, dinv, b2, z, N, DOUT);
    {
        long long threads = (long long)E * 32;
        k_scatter64<<<(int)((threads + T - 1) / T), T, 0, stream>>>(src, dst, hw, dinv, z, E);
    }

    // head
    {
        long long threads = (long long)N * 32;
        k_head<<<(int)((threads + T - 1) / T), T, 0, stream>>>(z, fcW, fcb, out, N);
    }
}